// AnchorProximityPE_15985868275718
// MI455X (gfx1250) — compile-verified
//
#include <hip/hip_runtime.h>
#include <stdint.h>

typedef __attribute__((ext_vector_type(2))) float v2f;
typedef __attribute__((ext_vector_type(8))) float v8f;

#define MAX_DIST 5
#define NBINS (MAX_DIST + 1)
#define D_PE 16

// ---------------------------------------------------------------------------
// Kernel 1: zero all BFS state (visited/frontier/next masks + histogram).
// ---------------------------------------------------------------------------
__global__ void zero_state(uint32_t* visLo, uint32_t* visHi,
                           uint32_t* froLo, uint32_t* froHi,
                           uint32_t* nxtLo, uint32_t* nxtHi,
                           uint32_t* hist, int* nvalid, int N) {
    int n = blockIdx.x * blockDim.x + threadIdx.x;
    if (n == 0) *nvalid = 0;
    if (n < N) {
        visLo[n] = 0u; visHi[n] = 0u;
        froLo[n] = 0u; froHi[n] = 0u;
        nxtLo[n] = 0u; nxtHi[n] = 0u;
        #pragma unroll
        for (int d = 0; d < NBINS; ++d) hist[(size_t)n * NBINS + d] = 0u;
    }
}

// ---------------------------------------------------------------------------
// Kernel 2: gather anchor entities, dedup (== jnp.unique semantics: only the
// set + count matter; bit order is irrelevant since counts are symmetric over
// sources), seed frontier/visited bits and the depth-0 histogram bin.
// One block of 64 threads.
// ---------------------------------------------------------------------------
__global__ void seed_sources(const int* __restrict__ h_ids,
                             const int* __restrict__ t_ids,
                             const int* __restrict__ anchors, int numAnchors,
                             uint32_t* visLo, uint32_t* visHi,
                             uint32_t* froLo, uint32_t* froHi,
                             uint32_t* hist, int* nvalid) {
    __shared__ int ents[64];
    int k = threadIdx.x;                     // 0..63
    int S = 2 * numAnchors; if (S > 64) S = 64;
    int e = -1;
    if (k < S) {
        int ti = anchors[(k < numAnchors) ? k : (k - numAnchors)];
        e = (k < numAnchors) ? h_ids[ti] : t_ids[ti];
    }
    ents[k] = e;
    __syncthreads();
    if (k < S) {
        bool unique = true;
        for (int j = 0; j < k; ++j) if (ents[j] == e) unique = false;
        if (unique) {
            atomicAdd(nvalid, 1);
            if (k < 32) {
                atomicOr(&visLo[e], 1u << k);
                atomicOr(&froLo[e], 1u << k);
            } else {
                atomicOr(&visHi[e], 1u << (k - 32));
                atomicOr(&froHi[e], 1u << (k - 32));
            }
            // unique sources are distinct entities -> plain store is safe
            hist[(size_t)e * NBINS + 0] = 1u;
        }
    }
}

// ---------------------------------------------------------------------------
// Kernel 3: one BFS hop — OR-scatter frontier masks along both edge
// directions.  Pre-check against `next` to skip already-subsumed atomics
// (cheap L2-cached read vs. an atomic RMW).
// ---------------------------------------------------------------------------
__global__ void scatter_frontier(const int* __restrict__ h_ids,
                                 const int* __restrict__ t_ids, int E,
                                 const uint32_t* __restrict__ froLo,
                                 const uint32_t* __restrict__ froHi,
                                 uint32_t* nxtLo, uint32_t* nxtHi) {
    int i = blockIdx.x * blockDim.x + threadIdx.x;
    if (i >= E) return;
    int h = h_ids[i], t = t_ids[i];

    uint32_t lo = froLo[h], hi = froHi[h];
    if (lo && (nxtLo[t] & lo) != lo) atomicOr(&nxtLo[t], lo);
    if (hi && (nxtHi[t] & hi) != hi) atomicOr(&nxtHi[t], hi);

    lo = froLo[t]; hi = froHi[t];
    if (lo && (nxtLo[h] & lo) != lo) atomicOr(&nxtLo[h], lo);
    if (hi && (nxtHi[h] & hi) != hi) atomicOr(&nxtHi[h], hi);
}

// ---------------------------------------------------------------------------
// Kernel 4: commit hop — newly = next & ~visited; record popcount into the
// depth bin; advance frontier; clear next for the following hop.
// ---------------------------------------------------------------------------
__global__ void advance_frontier(uint32_t* visLo, uint32_t* visHi,
                                 uint32_t* froLo, uint32_t* froHi,
                                 uint32_t* nxtLo, uint32_t* nxtHi,
                                 uint32_t* hist, int depth, int N) {
    int n = blockIdx.x * blockDim.x + threadIdx.x;
    if (n >= N) return;
    uint32_t nl = nxtLo[n], nh = nxtHi[n];
    uint32_t vl = visLo[n], vh = visHi[n];
    uint32_t newL = nl & ~vl, newH = nh & ~vh;
    visLo[n] = vl | newL; visHi[n] = vh | newH;
    froLo[n] = newL;      froHi[n] = newH;
    nxtLo[n] = 0u;        nxtHi[n] = 0u;
    hist[(size_t)n * NBINS + depth] = (uint32_t)(__popc(newL) + __popc(newH));
}

// ---------------------------------------------------------------------------
// Kernel 5: out = (hist / n_valid) @ dist_embed via V_WMMA_F32_16X16X4_F32.
// One wave32 per 16-entity tile.  K=6 padded to 8 -> two K=4 f32 WMMAs.
// A layout (32-bit 16x4, ISA 7.12.2): lanes 0-15 hold K=0,1 in v0,v1;
// lanes 16-31 hold K=2,3.  B/C/D: rows striped across lanes, M-halves split.
// The divide by n_valid is folded into B (exact integers stay in A).
// h5 = n_valid - sum(h0..h4): covers depth-5 AND unreachable (dist init = 5).
// ---------------------------------------------------------------------------
__global__ void finalize_wmma(const uint32_t* __restrict__ hist,
                              const int* __restrict__ nvalid,
                              const float* __restrict__ embed,
                              float* __restrict__ out, int N) {
    int wave = (int)((blockIdx.x * blockDim.x + threadIdx.x) >> 5);
    int lane = threadIdx.x & 31;
    int base = wave * 16;
    if (base >= N) return;              // wave-uniform: EXEC stays all-ones

    int half = lane >> 4;               // 0: K=0,1 / M rows 0..7 ; 1: K=2,3 / M rows 8..15
    int col  = lane & 15;
    float nv  = (float)(*nvalid);
    float inv = 1.0f / fmaxf(nv, 1.0f);

    // B operands: rows of (dist_embed * inv); second WMMA pads rows 6,7 -> 0
    v2f b0, b1;
    b0.x = embed[(half ? 2 : 0) * D_PE + col] * inv;
    b0.y = embed[(half ? 3 : 1) * D_PE + col] * inv;
    b1.x = half ? 0.0f : embed[4 * D_PE + col] * inv;
    b1.y = half ? 0.0f : embed[5 * D_PE + col] * inv;

    // A operands: 16-row histogram tile (exact small integers in f32)
    int row = base + col; if (row > N - 1) row = N - 1;   // clamp loads; stores guarded
    const uint32_t* hr = hist + (size_t)row * NBINS;
    float h0 = (float)hr[0], h1 = (float)hr[1], h2 = (float)hr[2];
    float h3 = (float)hr[3], h4 = (float)hr[4];
    float h5 = nv - (h0 + h1 + h2 + h3 + h4);

    v2f a0, a1;
    a0.x = half ? h2 : h0;
    a0.y = half ? h3 : h1;
    a1.x = half ? 0.0f : h4;            // K cols 4,5 live; 6,7 are zero pads
    a1.y = half ? 0.0f : h5;

    v8f c = {};
    c = __builtin_amdgcn_wmma_f32_16x16x4_f32(false, a0, false, b0, (short)0, c, false, false);
    c = __builtin_amdgcn_wmma_f32_16x16x4_f32(false, a1, false, b1, (short)0, c, false, false);

    // C/D layout: VGPR r -> row M = r (+8 for high half), N = col.
    if (base + 16 <= N) {
        // Full tile (always the case when N % 16 == 0): wave-uniform branch,
        // no per-store exec-mask juggling.
        float* o = out + ((size_t)(base + half * 8)) * D_PE + col;
        #pragma unroll
        for (int r = 0; r < 8; ++r) o[(size_t)r * D_PE] = c[r];
    } else {
        #pragma unroll
        for (int r = 0; r < 8; ++r) {
            int m = base + r + half * 8;
            if (m < N) out[(size_t)m * D_PE + col] = c[r];
        }
    }
}

// ---------------------------------------------------------------------------
// Host launcher
// ---------------------------------------------------------------------------
extern "C" void kernel_launch(void* const* d_in, const int* in_sizes, int n_in,
                              void* d_out, int out_size, void* d_ws, size_t ws_size,
                              hipStream_t stream) {
    const int*   h_ids   = (const int*)d_in[0];
    const int*   t_ids   = (const int*)d_in[1];
    const int*   anchors = (const int*)d_in[2];
    // d_in[3] = num_entities scalar (device); N derived from out_size instead.
    const float* embed   = (const float*)d_in[4];
    float*       out     = (float*)d_out;

    const int E = in_sizes[0];
    const int A = in_sizes[2];
    const int N = out_size / D_PE;

    // Workspace carve-up (all u32): 6*N masks + 6*N hist + 1 count  (~2.4 MB)
    uint32_t* w      = (uint32_t*)d_ws;
    uint32_t* visLo  = w;                 w += N;
    uint32_t* visHi  = w;                 w += N;
    uint32_t* froLo  = w;                 w += N;
    uint32_t* froHi  = w;                 w += N;
    uint32_t* nxtLo  = w;                 w += N;
    uint32_t* nxtHi  = w;                 w += N;
    uint32_t* hist   = w;                 w += (size_t)N * NBINS;
    int*      nvalid = (int*)w;

    const int TPB = 256;
    const int gN  = (N + TPB - 1) / TPB;
    const int gE  = (E + TPB - 1) / TPB;

    zero_state<<<gN, TPB, 0, stream>>>(visLo, visHi, froLo, froHi,
                                       nxtLo, nxtHi, hist, nvalid, N);

    seed_sources<<<1, 64, 0, stream>>>(h_ids, t_ids, anchors, A,
                                       visLo, visHi, froLo, froHi, hist, nvalid);

    for (int depth = 1; depth <= MAX_DIST; ++depth) {
        scatter_frontier<<<gE, TPB, 0, stream>>>(h_ids, t_ids, E,
                                                 froLo, froHi, nxtLo, nxtHi);
        advance_frontier<<<gN, TPB, 0, stream>>>(visLo, visHi, froLo, froHi,
                                                 nxtLo, nxtHi, hist, depth, N);
    }

    const int tiles  = (N + 15) / 16;
    const int wavesB = 8;                               // 256 threads = 8 waves
    const int blocks = (tiles + wavesB - 1) / wavesB;
    finalize_wmma<<<blocks, wavesB * 32, 0, stream>>>(hist, nvalid, embed, out, N);
}